// InputEmbeddingLayer_61040075211260
// MI455X (gfx1250) — compile-verified
//
#include <hip/hip_runtime.h>
#include <math.h>

typedef float v2f __attribute__((ext_vector_type(2)));
typedef float v8f __attribute__((ext_vector_type(8)));

#define NNODES 20000
#define NEDGES 200000

// LDS strides (floats). Kept even so v2f (8B) LDS loads stay aligned; odd/64
// residues chosen to avoid bank conflicts on column-wise access.
#define SSTR 162   // s buffer rows (16x... 32x160 data)
#define HSTR 66    // MLP hidden rows (32x64 data)
#define OSTR 34    // o0/q rows (32x32 data)

// D = A(16x4 f32) * B(4x16 f32) + C(16x16 f32), wave32.
__device__ __forceinline__ v8f wmma_f32(v2f a, v2f b, v8f c) {
  return __builtin_amdgcn_wmma_f32_16x16x4_f32(false, a, false, b, (short)0, c,
                                               false, false);
}

#define V8ZERO {0.f,0.f,0.f,0.f,0.f,0.f,0.f,0.f}

// Packed-weight B fragment: P[((k>>2)*N + n)*4 + (k&3)] = W[k][n].
// One aligned b64 load per WMMA, coalesced across lanes.
__device__ __forceinline__ v2f loadB(const float* __restrict__ P, int N,
                                     int k0, int n, int half) {
  return *(const v2f*)(P + ((size_t)(k0 >> 2) * N + n) * 4 + 2 * half);
}

// ---------------------------------------------------------------------------
// Weight packing prep kernels (run every launch; deterministic).
// ---------------------------------------------------------------------------
__global__ void pack_kernel(const float* __restrict__ src, float* __restrict__ dst,
                            int K, int N) {
  int idx = blockIdx.x * 256 + threadIdx.x;
  if (idx >= K * N) return;
  int k = idx / N, n = idx - k * N;
  dst[((size_t)(k >> 2) * N + n) * 4 + (k & 3)] = src[idx];
}

// Concatenate fuse_w0|w1|w2 (160 x [128|64|32]) into packed 160x224.
__global__ void pack_fuse_kernel(const float* __restrict__ w0,
                                 const float* __restrict__ w1,
                                 const float* __restrict__ w2,
                                 float* __restrict__ dst) {
  int idx = blockIdx.x * 256 + threadIdx.x;
  if (idx >= 160 * 224) return;
  int k = idx / 224, n = idx - k * 224;
  float v;
  if (n < 128)      v = w0[k * 128 + n];
  else if (n < 192) v = w1[k * 64 + (n - 128)];
  else              v = w2[k * 32 + (n - 192)];
  dst[((size_t)(k >> 2) * 224 + n) * 4 + (k & 3)] = v;
}

// ---------------------------------------------------------------------------
// Node kernel: node_scal + PE into d_out[:,:128], zero-fill cols 128..479,
// and materialize nf2 = ring_emb_ed[ring] + onehot @ atom_W_ed into workspace.
// ---------------------------------------------------------------------------
__global__ __launch_bounds__(128) void node_kernel(
    const float* __restrict__ atom_oh, const int* __restrict__ ring_info,
    const int* __restrict__ tarr,
    const float* __restrict__ ring_emb_node, const float* __restrict__ atom_W_node,
    const float* __restrict__ feat_W, const float* __restrict__ feat_b,
    const float* __restrict__ ring_emb_ed, const float* __restrict__ atom_W_ed,
    float* __restrict__ nf2, float* __restrict__ node_out)
{
  const int n   = blockIdx.x;
  const int tid = threadIdx.x;
  __shared__ float nf_sh[32];
  const int ring = ring_info[n];
  if (tid < 32) {
    float a1 = ring_emb_node[ring * 32 + tid];
    float a2 = ring_emb_ed[ring * 32 + tid];
    for (int a = 0; a < 6; ++a) {
      float o = atom_oh[n * 6 + a];
      a1 += o * atom_W_node[a * 32 + tid];
      a2 += o * atom_W_ed[a * 32 + tid];
    }
    nf_sh[tid] = a1;
    nf2[(size_t)n * 32 + tid] = a2;
  }
  __syncthreads();

  float acc = feat_b[tid];
  for (int k = 0; k < 32; ++k) acc += nf_sh[k] * feat_W[k * 128 + tid];

  float tn   = (float)tarr[n];
  float freq = expf(-9.210340371976184f * (float)(tid & ~1) * (1.0f / 128.0f));
  float ang  = tn * freq;
  float pe   = (tid & 1) ? cosf(ang) : sinf(ang);

  size_t base = (size_t)n * 480;
  node_out[base + tid]       = acc + pe;
  node_out[base + 128 + tid] = 0.f;
  node_out[base + 256 + tid] = 0.f;
  if (tid < 96) node_out[base + 384 + tid] = 0.f;
}

// LayerNorm(64)+SiLU over 32 rows (2 per lane: row, row+16); cols split by
// half, cross-half combine via shfl_xor(16).
__device__ __forceinline__ void ln_silu2(float* H, const float* __restrict__ g,
                                         const float* __restrict__ b,
                                         int row, int half) {
  const int cb = half * 32;
  float s0 = 0.f, q0 = 0.f, s1 = 0.f, q1 = 0.f;
  for (int j = 0; j < 32; ++j) {
    float x0 = H[row * HSTR + cb + j];        s0 += x0; q0 += x0 * x0;
    float x1 = H[(row + 16) * HSTR + cb + j]; s1 += x1; q1 += x1 * x1;
  }
  s0 += __shfl_xor(s0, 16); q0 += __shfl_xor(q0, 16);
  s1 += __shfl_xor(s1, 16); q1 += __shfl_xor(q1, 16);
  float m0 = s0 * (1.f / 64.f), v0 = q0 * (1.f / 64.f) - m0 * m0;
  float m1 = s1 * (1.f / 64.f), v1 = q1 * (1.f / 64.f) - m1 * m1;
  float r0 = rsqrtf(v0 + 1e-5f), r1 = rsqrtf(v1 + 1e-5f);
  for (int j = 0; j < 32; ++j) {
    int c = cb + j;
    float x = (H[row * HSTR + c] - m0) * r0 * g[c] + b[c];
    H[row * HSTR + c] = x / (1.f + expf(-x));
    float y = (H[(row + 16) * HSTR + c] - m1) * r1 * g[c] + b[c];
    H[(row + 16) * HSTR + c] = y / (1.f + expf(-y));
  }
}

// ---------------------------------------------------------------------------
// Edge kernel: 2 waves/block, each wave owns 32 edges = 2 WMMA M-tiles, so one
// B fragment (single b64 from packed weights) feeds two v_wmma ops.
// ---------------------------------------------------------------------------
__global__ __launch_bounds__(64) void edge_kernel(
    const float* __restrict__ pos, const float* __restrict__ bond_oh,
    const int* __restrict__ edge_index,
    const float* __restrict__ rbf_mean, const float* __restrict__ rbf_std,
    const float* __restrict__ rbf_w, const float* __restrict__ rbf_b,
    const float* __restrict__ bond_W,
    const float* __restrict__ Pfuse,
    const float* __restrict__ Prad1, const float* __restrict__ rad_b1,
    const float* __restrict__ ln1_g, const float* __restrict__ ln1_b,
    const float* __restrict__ Prad2, const float* __restrict__ rad_b2,
    const float* __restrict__ ln2_g, const float* __restrict__ ln2_b,
    const float* __restrict__ Prad3, const float* __restrict__ rad_off,
    const float* __restrict__ Pproj0, const float* __restrict__ proj_b0,
    const float* __restrict__ Pproj2,
    const float* __restrict__ nf2,
    float* __restrict__ node_out, float* __restrict__ fused_out)
{
  // Per-wave LDS. sBuf: s (32 x 160, stride SSTR). Reused later as:
  //   Wbuf = sBuf[0..2048)        (w, 32x64, stride 64)
  //   O0   = sBuf[2048..3136)     (o0, 32x32, stride OSTR)
  //   Q    = sBuf[3136..4224)     (q,  32x32, stride OSTR)
  __shared__ float sBuf[2][32 * SSTR];
  __shared__ float sHm[2][32 * HSTR];
  __shared__ float sSHm[2][32 * 9];
  __shared__ int   sIDXm[2][64];     // [0:32) src, [32:64) dst

  const int tid  = threadIdx.x;
  const int w    = tid >> 5;
  const int lane = tid & 31;
  const int row  = lane & 15;   // A row within M-tile / B,D column
  const int half = lane >> 4;
  const int col  = row;

  float* S   = sBuf[w];
  float* H   = sHm[w];
  float* SH  = sSHm[w];
  int*   IDX = sIDXm[w];

  const int e0 = (blockIdx.x * 2 + w) * 32;   // 6250 wave-jobs, exact
  const int e  = e0 + lane;                   // this lane's edge

  // ---- Stage A: geometry + spherical harmonics (one edge per lane) ----
  const int src = edge_index[e];
  const int dst = edge_index[NEDGES + e];
  float px = pos[src * 3 + 0] - pos[dst * 3 + 0];
  float py = pos[src * 3 + 1] - pos[dst * 3 + 1];
  float pz = pos[src * 3 + 2] - pos[dst * 3 + 2];
  float dist = sqrtf(px * px + py * py + pz * pz);
  float invn = 1.0f / fmaxf(dist, 1e-12f);
  float ux = px * invn, uy = py * invn, uz = pz * invn;
  const float s3 = 1.7320508075688772f, s5 = 2.23606797749979f,
              s15 = 3.872983346207417f;
  SH[lane * 9 + 0] = 1.0f;
  SH[lane * 9 + 1] = s3 * uy;
  SH[lane * 9 + 2] = s3 * uz;
  SH[lane * 9 + 3] = s3 * ux;
  SH[lane * 9 + 4] = s15 * ux * uy;
  SH[lane * 9 + 5] = s15 * uy * uz;
  SH[lane * 9 + 6] = 0.5f * s5 * (3.0f * uz * uz - 1.0f);
  SH[lane * 9 + 7] = s15 * ux * uz;
  SH[lane * 9 + 8] = 0.5f * s15 * (ux * ux - uy * uy);
  IDX[lane]      = src;
  IDX[32 + lane] = dst;

  // ---- Stage B: s = [rbf(128) | bond @ bond_W (32)], full row per lane ----
  const float rbfw = rbf_w[0], rbfb = rbf_b[0];
  const float xr = rbfw * (dist * 1e-3f) + rbfb;   // dist / CUTOFF
  for (int k = 0; k < 128; ++k) {
    float stdv = fabsf(rbf_std[k]) + 1e-5f;
    float dlt  = (xr - rbf_mean[k]) / stdv;
    S[lane * SSTR + k] = expf(-0.5f * dlt * dlt) * (0.39894244f / stdv);
  }
  {
    float b0 = bond_oh[(size_t)e * 5 + 0], b1 = bond_oh[(size_t)e * 5 + 1];
    float b2 = bond_oh[(size_t)e * 5 + 2], b3 = bond_oh[(size_t)e * 5 + 3];
    float b4 = bond_oh[(size_t)e * 5 + 4];
    for (int j = 0; j < 32; ++j)
      S[lane * SSTR + 128 + j] =
          b0 * bond_W[j] + b1 * bond_W[32 + j] + b2 * bond_W[64 + j] +
          b3 * bond_W[96 + j] + b4 * bond_W[128 + j];
  }
  __syncthreads();

  // ---- Stage C: fused = (s @ Wcat) * sh * inv ; K=160, N=224 ----
  const float invF = 0.07905694150420949f;  // 1/sqrt(160)
  for (int t = 0; t < 14; ++t) {
    const int n0 = t * 16;
    v8f acc0 = V8ZERO, acc1 = V8ZERO;
    for (int k0 = 0; k0 < 160; k0 += 4) {
      int ka = k0 + 2 * half;
      v2f b  = loadB(Pfuse, 224, k0, n0 + col, half);
      v2f a0 = *(const v2f*)&S[row * SSTR + ka];
      v2f a1 = *(const v2f*)&S[(row + 16) * SSTR + ka];
      acc0 = wmma_f32(a0, b, acc0);
      acc1 = wmma_f32(a1, b, acc1);
    }
    for (int m = 0; m < 2; ++m) {
      v8f acc = m ? acc1 : acc0;
      for (int r = 0; r < 8; ++r) {
        int Mr = r + 8 * half + 16 * m;
        size_t eb = (size_t)(e0 + Mr) * 480;
        float v = acc[r] * invF;
        if (n0 < 128) {
          fused_out[eb + n0 + col] = v * SH[Mr * 9 + 0];
        } else if (n0 < 192) {
          int u = n0 + col - 128;
          for (int d = 0; d < 3; ++d)
            fused_out[eb + 128 + u * 3 + d] = v * SH[Mr * 9 + 1 + d];
        } else {
          int u = n0 + col - 192;
          for (int d = 0; d < 5; ++d)
            fused_out[eb + 320 + u * 5 + d] = v * SH[Mr * 9 + 4 + d];
        }
      }
    }
  }

  // ---- Stage D1: h1 = silu(LN(rbf @ rad_W1 + b1)) ; K=128, N=64 ----
  {
    v8f hacc[4][2];
    for (int t = 0; t < 4; ++t) {
      v8f acc0 = V8ZERO, acc1 = V8ZERO;
      for (int k0 = 0; k0 < 128; k0 += 4) {
        int ka = k0 + 2 * half;
        v2f b  = loadB(Prad1, 64, k0, t * 16 + col, half);
        v2f a0 = *(const v2f*)&S[row * SSTR + ka];
        v2f a1 = *(const v2f*)&S[(row + 16) * SSTR + ka];
        acc0 = wmma_f32(a0, b, acc0);
        acc1 = wmma_f32(a1, b, acc1);
      }
      hacc[t][0] = acc0; hacc[t][1] = acc1;
    }
    for (int t = 0; t < 4; ++t)
      for (int r = 0; r < 8; ++r) {
        float bb = rad_b1[t * 16 + col];
        H[(r + 8 * half) * HSTR + t * 16 + col]      = hacc[t][0][r] + bb;
        H[(r + 8 * half + 16) * HSTR + t * 16 + col] = hacc[t][1][r] + bb;
      }
  }
  __syncthreads();
  ln_silu2(H, ln1_g, ln1_b, row, half);
  __syncthreads();

  // ---- Stage D2: h2 = silu(LN(h1 @ rad_W2 + b2)) ; K=64, N=64 ----
  {
    v8f hacc[4][2];
    for (int t = 0; t < 4; ++t) {
      v8f acc0 = V8ZERO, acc1 = V8ZERO;
      for (int k0 = 0; k0 < 64; k0 += 4) {
        int ka = k0 + 2 * half;
        v2f b  = loadB(Prad2, 64, k0, t * 16 + col, half);
        v2f a0 = *(const v2f*)&H[row * HSTR + ka];
        v2f a1 = *(const v2f*)&H[(row + 16) * HSTR + ka];
        acc0 = wmma_f32(a0, b, acc0);
        acc1 = wmma_f32(a1, b, acc1);
      }
      hacc[t][0] = acc0; hacc[t][1] = acc1;
    }
    __syncthreads();  // all reads of H done before overwrite
    for (int t = 0; t < 4; ++t)
      for (int r = 0; r < 8; ++r) {
        float bb = rad_b2[t * 16 + col];
        H[(r + 8 * half) * HSTR + t * 16 + col]      = hacc[t][0][r] + bb;
        H[(r + 8 * half + 16) * HSTR + t * 16 + col] = hacc[t][1][r] + bb;
      }
  }
  __syncthreads();
  ln_silu2(H, ln2_g, ln2_b, row, half);
  __syncthreads();

  // ---- Stage D3: w = h2 @ rad_W3 + offset ; K=64, N=64 -> Wbuf (reuse S) ----
  float* Wbuf = S;          // 32x64, stride 64
  float* O0   = S + 2048;   // 32x32, stride OSTR
  float* Q    = S + 3136;   // 32x32, stride OSTR
  {
    v8f hacc[4][2];
    for (int t = 0; t < 4; ++t) {
      v8f acc0 = V8ZERO, acc1 = V8ZERO;
      for (int k0 = 0; k0 < 64; k0 += 4) {
        int ka = k0 + 2 * half;
        v2f b  = loadB(Prad3, 64, k0, t * 16 + col, half);
        v2f a0 = *(const v2f*)&H[row * HSTR + ka];
        v2f a1 = *(const v2f*)&H[(row + 16) * HSTR + ka];
        acc0 = wmma_f32(a0, b, acc0);
        acc1 = wmma_f32(a1, b, acc1);
      }
      hacc[t][0] = acc0; hacc[t][1] = acc1;
    }
    __syncthreads();  // S region may still be read as A by other lanes? (h1 done long ago; keep for safety vs Wbuf aliasing)
    for (int t = 0; t < 4; ++t)
      for (int r = 0; r < 8; ++r) {
        float bb = rad_off[t * 16 + col];
        Wbuf[(r + 8 * half) * 64 + t * 16 + col]      = hacc[t][0][r] + bb;
        Wbuf[(r + 8 * half + 16) * 64 + t * 16 + col] = hacc[t][1][r] + bb;
      }
  }
  __syncthreads();

  // ---- Stage E: o0 = w[:32]*xs*sh0 ; q = w[32:]*xs (full row per lane) ----
  {
    int srcR  = IDX[lane];
    float sh0 = SH[lane * 9 + 0];
    for (int u = 0; u < 32; ++u) {
      float xs = nf2[(size_t)srcR * 32 + u];
      O0[lane * OSTR + u] = Wbuf[lane * 64 + u] * xs * sh0;
      Q[lane * OSTR + u]  = Wbuf[lane * 64 + 32 + u] * xs;
    }
  }
  __syncthreads();

  const float invs = 0.31622776601683794f;  // 1/sqrt(AVG_DEG)

  // ---- Stage F: p0 = o0 @ proj_W0 + b0, scatter into node cols 0..127 ----
  for (int t = 0; t < 8; ++t) {
    v8f acc0 = V8ZERO, acc1 = V8ZERO;
    for (int k0 = 0; k0 < 32; k0 += 4) {
      int ka = k0 + 2 * half;
      v2f b  = loadB(Pproj0, 128, k0, t * 16 + col, half);
      v2f a0 = *(const v2f*)&O0[row * OSTR + ka];
      v2f a1 = *(const v2f*)&O0[(row + 16) * OSTR + ka];
      acc0 = wmma_f32(a0, b, acc0);
      acc1 = wmma_f32(a1, b, acc1);
    }
    int n = t * 16 + col;
    float pb = proj_b0[n];
    for (int m = 0; m < 2; ++m) {
      v8f acc = m ? acc1 : acc0;
      for (int r = 0; r < 8; ++r) {
        int Mr = r + 8 * half + 16 * m;
        int dn = IDX[32 + Mr];
        atomicAdd(&node_out[(size_t)dn * 480 + n], (acc[r] + pb) * invs);
      }
    }
  }

  // ---- Stage G: p2 = (q @ proj_W2) * sh2, scatter into node cols 320..479 ----
  for (int t = 0; t < 2; ++t) {
    v8f acc0 = V8ZERO, acc1 = V8ZERO;
    for (int k0 = 0; k0 < 32; k0 += 4) {
      int ka = k0 + 2 * half;
      v2f b  = loadB(Pproj2, 32, k0, t * 16 + col, half);
      v2f a0 = *(const v2f*)&Q[row * OSTR + ka];
      v2f a1 = *(const v2f*)&Q[(row + 16) * OSTR + ka];
      acc0 = wmma_f32(a0, b, acc0);
      acc1 = wmma_f32(a1, b, acc1);
    }
    int v = t * 16 + col;
    for (int m = 0; m < 2; ++m) {
      v8f acc = m ? acc1 : acc0;
      for (int r = 0; r < 8; ++r) {
        int Mr = r + 8 * half + 16 * m;
        int dn = IDX[32 + Mr];
        for (int mm = 0; mm < 5; ++mm)
          atomicAdd(&node_out[(size_t)dn * 480 + 320 + v * 5 + mm],
                    acc[r] * SH[Mr * 9 + 4 + mm] * invs);
      }
    }
  }
}

extern "C" void kernel_launch(void* const* d_in, const int* in_sizes, int n_in,
                              void* d_out, int out_size, void* d_ws, size_t ws_size,
                              hipStream_t stream) {
  const float* atom_oh       = (const float*)d_in[0];
  const float* pos           = (const float*)d_in[1];
  const float* bond_oh       = (const float*)d_in[2];
  const int*   ring_info     = (const int*)d_in[3];
  const int*   edge_index    = (const int*)d_in[4];
  /* d_in[5] = batch (unused) */
  const int*   tarr          = (const int*)d_in[6];
  const float* ring_emb_node = (const float*)d_in[7];
  const float* atom_W_node   = (const float*)d_in[8];
  const float* feat_W        = (const float*)d_in[9];
  const float* feat_b        = (const float*)d_in[10];
  const float* rbf_mean      = (const float*)d_in[11];
  const float* rbf_std       = (const float*)d_in[12];
  const float* rbf_w         = (const float*)d_in[13];
  const float* rbf_b         = (const float*)d_in[14];
  const float* bond_W        = (const float*)d_in[15];
  const float* fuse_w0       = (const float*)d_in[16];
  const float* fuse_w1       = (const float*)d_in[17];
  const float* fuse_w2       = (const float*)d_in[18];
  const float* ring_emb_ed   = (const float*)d_in[19];
  const float* atom_W_ed     = (const float*)d_in[20];
  const float* rad_W1        = (const float*)d_in[21];
  const float* rad_b1        = (const float*)d_in[22];
  const float* ln1_g         = (const float*)d_in[23];
  const float* ln1_b         = (const float*)d_in[24];
  const float* rad_W2        = (const float*)d_in[25];
  const float* rad_b2        = (const float*)d_in[26];
  const float* ln2_g         = (const float*)d_in[27];
  const float* ln2_b         = (const float*)d_in[28];
  const float* rad_W3        = (const float*)d_in[29];
  const float* rad_off       = (const float*)d_in[30];
  const float* proj_W0       = (const float*)d_in[31];
  const float* proj_b0       = (const float*)d_in[32];
  const float* proj_W2       = (const float*)d_in[33];

  float* node_out  = (float*)d_out;
  float* fused_out = node_out + (size_t)NNODES * 480;

  // Workspace layout (floats)
  float* nf2    = (float*)d_ws;               // N*32      = 640000
  float* Pfuse  = nf2    + (size_t)NNODES*32; // 160*224   = 35840
  float* Prad1  = Pfuse  + 160 * 224;         // 128*64    = 8192
  float* Prad2  = Prad1  + 128 * 64;          // 64*64     = 4096
  float* Prad3  = Prad2  + 64 * 64;           // 64*64     = 4096
  float* Pproj0 = Prad3  + 64 * 64;           // 32*128    = 4096
  float* Pproj2 = Pproj0 + 32 * 128;          // 32*32     = 1024

  // Pack weights into WMMA-friendly (b64-per-fragment, coalesced) layout.
  pack_fuse_kernel<<<(160 * 224 + 255) / 256, 256, 0, stream>>>(fuse_w0, fuse_w1,
                                                                fuse_w2, Pfuse);
  pack_kernel<<<(128 * 64 + 255) / 256, 256, 0, stream>>>(rad_W1, Prad1, 128, 64);
  pack_kernel<<<(64 * 64 + 255) / 256, 256, 0, stream>>>(rad_W2, Prad2, 64, 64);
  pack_kernel<<<(64 * 64 + 255) / 256, 256, 0, stream>>>(rad_W3, Prad3, 64, 64);
  pack_kernel<<<(32 * 128 + 255) / 256, 256, 0, stream>>>(proj_W0, Pproj0, 32, 128);
  pack_kernel<<<(32 * 32 + 255) / 256, 256, 0, stream>>>(proj_W2, Pproj2, 32, 32);

  node_kernel<<<NNODES, 128, 0, stream>>>(
      atom_oh, ring_info, tarr, ring_emb_node, atom_W_node, feat_W, feat_b,
      ring_emb_ed, atom_W_ed, nf2, node_out);

  // 200000 edges / (2 waves * 32 edges) = 3125 blocks, no tail.
  edge_kernel<<<NEDGES / 64, 64, 0, stream>>>(
      pos, bond_oh, edge_index, rbf_mean, rbf_std, rbf_w, rbf_b, bond_W,
      Pfuse, Prad1, rad_b1, ln1_g, ln1_b, Prad2, rad_b2, ln2_g, ln2_b,
      Prad3, rad_off, Pproj0, proj_b0, Pproj2, nf2, node_out, fused_out);
}